// EKFGradient_1013612281886
// MI455X (gfx1250) — compile-verified
//
#include <hip/hip_runtime.h>
#include <stdint.h>

#ifndef __has_builtin
#define __has_builtin(x) 0
#endif

#define T_LEN   2048
#define CHUNK   64
#define NCHUNK  (T_LEN / CHUNK)        // 32 chunks per trajectory
#define ROWS    32                     // trajectories per block == wave32
#define ROW_DW  (CHUNK * 3)            // 192 dwords of z data per row
#define PROW_DW 195                    // 192 + 3 pad dwords (1 per 64) -> bank-conflict-free
#define BUF_DW  (ROWS * PROW_DW)       // 6240 dwords per buffer
#define DT      (1.0f / 120.0f)

#define HAVE_TDM __has_builtin(__builtin_amdgcn_tensor_load_to_lds)

typedef uint32_t u32x4 __attribute__((ext_vector_type(4)));
typedef int      i32x4 __attribute__((ext_vector_type(4)));
typedef int      i32x8 __attribute__((ext_vector_type(8)));
typedef float    v2f   __attribute__((ext_vector_type(2)));
typedef float    v8f   __attribute__((ext_vector_type(8)));

// ---------------------------------------------------------------------------
// TDM: one instruction loads a [32 rows x 192 dwords] tile from the strided
// measurement tensor (row stride = T*3 floats) into LDS, inserting 1 dword of
// padding every 64 dwords so the LDS row stride is 195 dwords (gcd(3,64)=1 ->
// all 32 lanes hit distinct banks).
// ---------------------------------------------------------------------------
__device__ __forceinline__ void tdm_load_tile(uint32_t lds_byte_addr, uint64_t gaddr) {
#if HAVE_TDM
  u32x4 g0;
  g0[0] = 1u;                                            // count=1 valid descriptor
  g0[1] = lds_byte_addr;                                 // lds_addr
  g0[2] = (uint32_t)(gaddr & 0xFFFFFFFFull);             // global_addr[31:0]
  g0[3] = (uint32_t)((gaddr >> 32) & 0x1FFFFFFull)       // global_addr[56:32]
        | 0x80000000u;                                   // type = 2 ("image")
  i32x8 g1;
  g1[0] = (int)((2u << 16)        // data_size = 4 bytes
              | (1u << 20)        // pad_enable
              | (5u << 22));      // pad_interval code 5 = every 64 DW; pad_amount 0 = 1 DW
  g1[1] = (int)((6144u & 0xFFFFu) << 16);   // tensor_dim0 lo = T*3 = 6144
  g1[2] = (int)((2048u & 0xFFFFu) << 16);   // tensor_dim0 hi=0 | tensor_dim1 lo = 2048
  g1[3] = (int)(192u << 16);                // tensor_dim1 hi=0 | tile_dim0 = 192 elements
  g1[4] = (int)ROWS;                        // tile_dim1 = 32 rows | tile_dim2 = 0
  g1[5] = (int)6144;                        // tensor_dim0_stride lo = 6144 elements
  g1[6] = 0;
  g1[7] = 0;
  i32x4 z4 = {0, 0, 0, 0};
#if defined(__clang_major__) && (__clang_major__ >= 23)
  i32x8 z8 = {0, 0, 0, 0, 0, 0, 0, 0};
  __builtin_amdgcn_tensor_load_to_lds(g0, g1, z4, z4, z8, 0);
#else
  __builtin_amdgcn_tensor_load_to_lds(g0, g1, z4, z4, 0);
#endif
#else
  (void)lds_byte_addr; (void)gaddr;
#endif
}

__device__ __forceinline__ void wait_tensorcnt_le1() {
#if __has_builtin(__builtin_amdgcn_s_wait_tensorcnt)
  __builtin_amdgcn_s_wait_tensorcnt(1);
#else
  asm volatile("s_wait_tensorcnt 0x1" ::: "memory");
#endif
}

// padded LDS index for row `lane`, dword j (0..191) within the row
__device__ __forceinline__ int lds_idx(int lane, int j) {
  return lane * PROW_DW + j + (j >> 6);
}

// upper-triangle index for symmetric 6x6 (21 entries)
__device__ __forceinline__ constexpr int UI(int r, int c) {
  return (r <= c) ? (r * 6 - (r * (r + 1)) / 2 + c)
                  : (c * 6 - (c * (c + 1)) / 2 + r);
}

// ---------------------------------------------------------------------------
// One EKF step per lane. P kept as 21-entry symmetric upper triangle: the
// exact update preserves symmetry (S symmetric, Pc = Pr^T), and this cuts the
// issue-bound VALU work ~25% (36->21 in F P F^T, 108->63 FMA in the downdate).
// ---------------------------------------------------------------------------
__device__ __forceinline__ void ekf_step(float s[6], float P[21], float& acc,
                                         float zx, float zy, float zt,
                                         float damping, float friction, float a) {
  const float dt = DT;
  // predict_state (sign(0)==0 as in jnp.sign)
  float sgx = (s[2] > 0.f) ? 1.f : ((s[2] < 0.f) ? -1.f : 0.f);
  float sgy = (s[3] > 0.f) ? 1.f : ((s[3] < 0.f) ? -1.f : 0.f);
  float sp0 = s[0] + s[2] * dt;
  float sp1 = s[1] + s[3] * dt;
  float sp2 = s[2] - dt * (damping * s[2] + friction * sgx);
  float sp3 = s[3] - dt * (damping * s[3] + friction * sgy);
  float sp4 = s[4] + s[5] * dt;
  float sp5 = s[5];

  // M = F P (full 6x6, symmetric reads of P)
  float M[36];
#pragma unroll
  for (int c = 0; c < 6; ++c) {
    M[0 * 6 + c] = P[UI(0, c)] + dt * P[UI(2, c)];
    M[1 * 6 + c] = P[UI(1, c)] + dt * P[UI(3, c)];
    M[2 * 6 + c] = a * P[UI(2, c)];
    M[3 * 6 + c] = a * P[UI(3, c)];
    M[4 * 6 + c] = P[UI(4, c)] + dt * P[UI(5, c)];
    M[5 * 6 + c] = P[UI(5, c)];
  }
  // Pp = M F^T + Q, upper triangle only (dead cols eliminated by DCE)
  float Pp[21];
#pragma unroll
  for (int r = 0; r < 6; ++r) {
    float col[6];
    col[0] = M[r * 6 + 0] + dt * M[r * 6 + 2];
    col[1] = M[r * 6 + 1] + dt * M[r * 6 + 3];
    col[2] = a * M[r * 6 + 2];
    col[3] = a * M[r * 6 + 3];
    col[4] = M[r * 6 + 4] + dt * M[r * 6 + 5];
    col[5] = M[r * 6 + 5];
#pragma unroll
    for (int c = r; c < 6; ++c) Pp[UI(r, c)] = col[c];
  }
  Pp[UI(0, 0)] += 2e-10f;  Pp[UI(1, 1)] += 2e-10f;  Pp[UI(2, 2)] += 3e-7f;
  Pp[UI(3, 3)] += 3e-7f;   Pp[UI(4, 4)] += 1e-2f;   Pp[UI(5, 5)] += 1e-1f;

  // S = H Pp H^T + R (rows/cols 0,1,4), symmetric
  float s00 = Pp[UI(0, 0)] + 2.5e-7f;
  float s01 = Pp[UI(0, 1)];
  float s02 = Pp[UI(0, 4)];
  float s11 = Pp[UI(1, 1)] + 2.5e-7f;
  float s12 = Pp[UI(1, 4)];
  float s22 = Pp[UI(4, 4)] + 9.1e-3f;

  // symmetric 3x3 inverse via cofactors (6 unique) + det
  float c00 = s11 * s22 - s12 * s12;
  float c01 = s02 * s12 - s01 * s22;
  float c02 = s01 * s12 - s02 * s11;
  float det = s00 * c00 + s01 * c01 + s02 * c02;
  float id  = 1.0f / det;
  float i00 = c00 * id;
  float i01 = c01 * id;
  float i02 = c02 * id;
  float i11 = (s00 * s22 - s02 * s02) * id;
  float i12 = (s02 * s01 - s00 * s12) * id;
  float i22 = (s00 * s11 - s01 * s01) * id;

  float y0 = zx - sp0, y1 = zy - sp1, y2 = zt - sp4;
  float w0 = i00 * y0 + i01 * y1 + i02 * y2;
  float w1 = i01 * y0 + i11 * y1 + i12 * y2;
  float w2 = i02 * y0 + i12 * y1 + i22 * y2;

  acc += det + (y0 * w0 + y1 * w1 + y2 * w2);

  // state update via s += Pc * (S^{-1} y)  (doesn't wait on K)
  // and K = Pc * S^{-1} for the covariance downdate
  float K0[6], K1[6], K2[6];
#pragma unroll
  for (int r = 0; r < 6; ++r) {
    float p0 = Pp[UI(r, 0)], p1 = Pp[UI(r, 1)], p2 = Pp[UI(r, 4)];
    float sr = p0 * w0 + p1 * w1 + p2 * w2;
    if (r == 0) s[0] = sp0 + sr;
    if (r == 1) s[1] = sp1 + sr;
    if (r == 2) s[2] = sp2 + sr;
    if (r == 3) s[3] = sp3 + sr;
    if (r == 4) s[4] = sp4 + sr;
    if (r == 5) s[5] = sp5 + sr;
    K0[r] = p0 * i00 + p1 * i01 + p2 * i02;
    K1[r] = p0 * i01 + p1 * i11 + p2 * i12;
    K2[r] = p0 * i02 + p1 * i12 + p2 * i22;
  }

  // P = Pp - K (H Pp): symmetric rank-3 downdate, upper triangle only
#pragma unroll
  for (int r = 0; r < 6; ++r) {
#pragma unroll
    for (int c = r; c < 6; ++c) {
      P[UI(r, c)] = Pp[UI(r, c)] -
                    (K0[r] * Pp[UI(0, c)] + K1[r] * Pp[UI(1, c)] + K2[r] * Pp[UI(4, c)]);
    }
  }
}

// ---------------------------------------------------------------------------
// Main kernel: 1 wave32 per block, 1 trajectory per lane, TDM double buffering.
// ---------------------------------------------------------------------------
__global__ void __launch_bounds__(ROWS)
ekf_scan_kernel(const float* __restrict__ params,
                const float* __restrict__ meas,
                float* __restrict__ partials) {
  __shared__ float sm[2 * BUF_DW];

  const int lane = threadIdx.x;                       // 0..31
  const uint64_t gbase =
      (uint64_t)(uintptr_t)meas +
      (uint64_t)(blockIdx.x * ROWS) * (uint64_t)(T_LEN * 3) * 4ull;

  const float friction = params[0];
  const float damping  = params[1];
  const float a = 1.0f - DT * damping;

  const uint32_t ldsA = (uint32_t)(uintptr_t)(&sm[0]);
  const uint32_t ldsB = (uint32_t)(uintptr_t)(&sm[BUF_DW]);

#if HAVE_TDM
  tdm_load_tile(ldsA, gbase);
  tdm_load_tile(ldsB, gbase + (uint64_t)(CHUNK * 3 * 4));
  wait_tensorcnt_le1();                               // chunk 0 resident
#else
  for (int i = lane; i < ROWS * ROW_DW; i += ROWS) {
    int row = i / ROW_DW, j = i - row * ROW_DW;
    sm[row * PROW_DW + j + (j >> 6)] =
        meas[(uint64_t)(blockIdx.x * ROWS + row) * (T_LEN * 3) + j];
  }
  __syncthreads();
#endif

  // init state from z[0], z[1]
  float s[6], P[21];
  {
    float z00 = sm[lds_idx(lane, 0)], z01 = sm[lds_idx(lane, 1)], z02 = sm[lds_idx(lane, 2)];
    float z10 = sm[lds_idx(lane, 3)], z11 = sm[lds_idx(lane, 4)], z12 = sm[lds_idx(lane, 5)];
    s[0] = z00; s[1] = z01; s[4] = z02;
    s[2] = (z10 - z00) / DT; s[3] = (z11 - z01) / DT; s[5] = (z12 - z02) / DT;
#pragma unroll
    for (int i = 0; i < 21; ++i) P[i] = 0.0f;
#pragma unroll
    for (int i = 0; i < 6; ++i) P[UI(i, i)] = 0.01f;
  }

  float acc = 0.0f;

#pragma unroll 1
  for (int c = 0; c < NCHUNK; ++c) {
#if HAVE_TDM
    if (c > 0) wait_tensorcnt_le1();                  // chunk c resident (in-order TDM) except c=NCHUNK-1: 1 load left, wait(1) passes
#else
    if (c > 0) {
      __syncthreads();
      float* dst = &sm[(c & 1) * BUF_DW];
      const float* src = meas + (uint64_t)(blockIdx.x * ROWS) * (T_LEN * 3) + c * ROW_DW;
      for (int i = lane; i < ROWS * ROW_DW; i += ROWS) {
        int row = i / ROW_DW, j = i - row * ROW_DW;
        dst[row * PROW_DW + j + (j >> 6)] = src[(uint64_t)row * (T_LEN * 3) + j];
      }
      __syncthreads();
    }
#endif
    const float* buf = &sm[(c & 1) * BUF_DW];
    const int tl0 = (c == 0) ? 1 : 0;                 // step t=0 feeds only the init
#pragma unroll 1
    for (int tl = tl0; tl < CHUNK; ++tl) {
      const int j = tl * 3;
      float zx = buf[lds_idx(lane, j)];
      float zy = buf[lds_idx(lane, j + 1)];
      float zt = buf[lds_idx(lane, j + 2)];
      ekf_step(s, P, acc, zx, zy, zt, damping, friction, a);
    }
#if HAVE_TDM
    if (c + 2 < NCHUNK) {                             // refill the buffer we just drained
      tdm_load_tile((c & 1) ? ldsB : ldsA,
                    gbase + (uint64_t)((c + 2) * CHUNK * 3 * 4));
    }
#endif
  }

  // deterministic wave reduction: per-trajectory mean, then block sum
  float v = acc * (1.0f / (float)(T_LEN - 1));
#pragma unroll
  for (int o = 16; o > 0; o >>= 1) v += __shfl_xor(v, o, 32);
  if (lane == 0) partials[blockIdx.x] = v;
}

// ---------------------------------------------------------------------------
// Final reduction of the 64 block partials as a degenerate GEMM:
// total = 1^T p  via one V_WMMA_F32_16X16X4_F32 (A = 64 partials as 16x4,
// B = all-ones). Since B is all-ones, any bijective placement of the 64
// values into A yields the same total, so the result is layout-robust and
// bitwise deterministic across replays.
// ---------------------------------------------------------------------------
__global__ void __launch_bounds__(32)
ekf_reduce_wmma(const float* __restrict__ partials, float* __restrict__ out,
                float scale) {
  const int lane = threadIdx.x;          // 0..31, EXEC all-ones for WMMA
#if __has_builtin(__builtin_amdgcn_wmma_f32_16x16x4_f32)
  const int m = lane & 15;
  const int kbase = (lane < 16) ? 0 : 2;
  v2f A;
  A.x = partials[4 * m + kbase + 0];
  A.y = partials[4 * m + kbase + 1];
  v2f B; B.x = 1.0f; B.y = 1.0f;
  v8f C = {};
  C = __builtin_amdgcn_wmma_f32_16x16x4_f32(false, A, false, B, (short)0, C,
                                            false, false);
  float t = ((C[0] + C[1]) + (C[2] + C[3])) + ((C[4] + C[5]) + (C[6] + C[7]));
  t += __shfl_xor(t, 16, 32);            // rows 0..7 + rows 8..15
#else
  float t = partials[2 * lane] + partials[2 * lane + 1];
#pragma unroll
  for (int o = 16; o > 0; o >>= 1) t += __shfl_xor(t, o, 32);
#endif
  if (lane == 0) out[0] = t * scale;
}

__global__ void ekf_reduce_serial(const float* __restrict__ partials,
                                  float* __restrict__ out, int nblk, float scale) {
  if (threadIdx.x == 0 && blockIdx.x == 0) {
    float sum = 0.0f;
    for (int i = 0; i < nblk; ++i) sum += partials[i];
    out[0] = sum * scale;
  }
}

extern "C" void kernel_launch(void* const* d_in, const int* in_sizes, int n_in,
                              void* d_out, int out_size, void* d_ws, size_t ws_size,
                              hipStream_t stream) {
  const float* params = (const float*)d_in[0];
  const float* meas   = (const float*)d_in[1];
  const int B    = in_sizes[1] / (T_LEN * 3);   // 2048
  const int nblk = B / ROWS;                    // 64 blocks x 1 wave
  float* partials = (float*)d_ws;

  ekf_scan_kernel<<<nblk, ROWS, 0, stream>>>(params, meas, partials);
  if (nblk == 64) {
    ekf_reduce_wmma<<<1, 32, 0, stream>>>(partials, (float*)d_out,
                                          1.0f / (float)B);
  } else {
    ekf_reduce_serial<<<1, 1, 0, stream>>>(partials, (float*)d_out, nblk,
                                           1.0f / (float)B);
  }
}